// TF_opt_model_38036230373772
// MI455X (gfx1250) — compile-verified
//
#include <hip/hip_runtime.h>
#include <math.h>

typedef __attribute__((ext_vector_type(2))) float v2f;
typedef __attribute__((ext_vector_type(8))) float v8f;

#define V_DIM 96
#define K_DIM 20
#define VK    1920          // V_DIM*K_DIM
#define ENC   16            // E*N*C = 4*2*2
#define B_DIM 256
#define D_DIM 1024
#define IJ    9

// workspace layout (in floats)
#define WS_LOGT 0           // 9*1920   = 17280   logT[ij][v*20+k]
#define WS_LQ   17280       // 144*20   = 2880    Lq[ij*16+enc][k]
#define WS_LMR  20160       // 9*256    = 2304    Lmr[ij][b]
#define WS_W    22464       // 144*1920 = 276480  W[ijenc][c]

// ---------------------------------------------------------------------------
// Zero the scalar output (d_out is poisoned by the harness).
__global__ void zero_out(float* out) {
    if (threadIdx.x == 0) out[0] = 0.0f;
}

// ---------------------------------------------------------------------------
// Small-parameter preprocessing: one wave, thread k owns column k (k < 20).
// Produces logT (9*1920) and Lq (144*20). All per-column, no cross-thread use.
__global__ void prep_small(const float* __restrict__ T0in, const float* __restrict__ tin,
                           const float* __restrict__ rin, const float* __restrict__ ein,
                           const float* __restrict__ nin, const float* __restrict__ cin,
                           float* __restrict__ ws) {
    __shared__ float sT0[V_DIM * K_DIM];
    __shared__ float slt[V_DIM * K_DIM];
    __shared__ float slr[V_DIM * K_DIM];

    const int k = threadIdx.x;
    if (k >= K_DIM) return;

    // softmax over V for column k of _T0, sigmoids of _t, _r
    float mx = -1e30f;
    for (int v = 0; v < V_DIM; ++v) mx = fmaxf(mx, T0in[v * K_DIM + k]);
    float ssum = 0.0f;
    for (int v = 0; v < V_DIM; ++v) {
        float e = __expf(T0in[v * K_DIM + k] - mx);
        sT0[v * K_DIM + k] = e;
        ssum += e;
    }
    float inv = 1.0f / ssum;
    for (int v = 0; v < V_DIM; ++v) {
        sT0[v * K_DIM + k] *= inv;
        slt[v * K_DIM + k] = 1.0f / (1.0f + __expf(-tin[v * K_DIM + k]));
        slr[v * K_DIM + k] = 1.0f / (1.0f + __expf(-rin[v * K_DIM + k]));
    }

    // marginal strand probabilities
    float pt = 0.0f, pr = 0.0f;
    for (int v = 0; v < V_DIM; ++v) {
        pt += sT0[v * K_DIM + k] * slt[v * K_DIM + k];
        pr += sT0[v * K_DIM + k] * slr[v * K_DIM + k];
    }

    // logT[ij][v*20+k] = log(T0*wt_i*wr_j) - log(norm_ij)
    float* logT = ws + WS_LOGT;
    for (int i = 0; i < 3; ++i) {
        for (int j = 0; j < 3; ++j) {
            float nrm = 0.0f;
            for (int v = 0; v < V_DIM; ++v) {
                float lt = slt[v * K_DIM + k], lr = slr[v * K_DIM + k];
                float wt = (i == 0) ? lt : ((i == 1) ? (1.0f - lt) : 1.0f);
                float wr = (j == 0) ? lr : ((j == 1) ? (1.0f - lr) : 1.0f);
                nrm += sT0[v * K_DIM + k] * wt * wr;
            }
            float lnrm = __logf(nrm);
            for (int v = 0; v < V_DIM; ++v) {
                float lt = slt[v * K_DIM + k], lr = slr[v * K_DIM + k];
                float wt = (i == 0) ? lt : ((i == 1) ? (1.0f - lt) : 1.0f);
                float wr = (j == 0) ? lr : ((j == 1) ? (1.0f - lr) : 1.0f);
                float val = sT0[v * K_DIM + k] * wt * wr;
                logT[(i * 3 + j) * VK + v * K_DIM + k] = __logf(val) - lnrm;
            }
        }
    }

    // log-softmax of context factors (per column k)
    float Lpe[4], Lpn[2], Lpc[2];
    {
        float m = -1e30f;
        for (int e = 0; e < 4; ++e) m = fmaxf(m, ein[e * K_DIM + k]);
        float s = 0.0f;
        for (int e = 0; e < 4; ++e) s += __expf(ein[e * K_DIM + k] - m);
        float l = m + __logf(s);
        for (int e = 0; e < 4; ++e) Lpe[e] = ein[e * K_DIM + k] - l;
    }
    {
        float m = fmaxf(nin[k], nin[K_DIM + k]);
        float s = __expf(nin[k] - m) + __expf(nin[K_DIM + k] - m);
        float l = m + __logf(s);
        Lpn[0] = nin[k] - l; Lpn[1] = nin[K_DIM + k] - l;
    }
    {
        float m = fmaxf(cin[k], cin[K_DIM + k]);
        float s = __expf(cin[k] - m) + __expf(cin[K_DIM + k] - m);
        float l = m + __logf(s);
        Lpc[0] = cin[k] - l; Lpc[1] = cin[K_DIM + k] - l;
    }

    const float Lqt[3] = { __logf(pt), __logf(1.0f - pt), 0.0f };
    const float Lqr[3] = { __logf(pr), __logf(1.0f - pr), 0.0f };

    float* Lq = ws + WS_LQ;
    for (int ij = 0; ij < IJ; ++ij) {
        int i = ij / 3, j = ij % 3;
        for (int enc = 0; enc < ENC; ++enc) {
            int e = enc >> 2, n = (enc >> 1) & 1, cc = enc & 1;
            Lq[(ij * ENC + enc) * K_DIM + k] =
                Lqt[i] + Lqr[j] + Lpe[e] + Lpn[n] + Lpc[cc];
        }
    }
}

// ---------------------------------------------------------------------------
// Missing-rate at the B sampled indices: one block per b.
// Lmr[ij][b] = log(mr[(i==2)][(j==2)]) from y_tr sums of Y at index[b].
__global__ void prep_mr(const float* __restrict__ Y, const int* __restrict__ index,
                        float* __restrict__ ws) {
    __shared__ float red[256];
    __shared__ float ytr[IJ];
    const int b = blockIdx.x;
    const int t = threadIdx.x;
    const int idx = index[b];

    for (int ij = 0; ij < IJ; ++ij) {
        float s = 0.0f;
        // Y[ij, enc, idx, v]: 16*96 = 1536 contiguous-per-v elements
        for (int e = t; e < ENC * V_DIM; e += 256) {
            int enc = e / V_DIM, v = e - enc * V_DIM;
            size_t base = ((size_t)(ij * ENC + enc) * D_DIM + (size_t)idx) * V_DIM;
            s += Y[base + v];
        }
        red[t] = s;
        __syncthreads();
        for (int o = 128; o > 0; o >>= 1) {
            if (t < o) red[t] += red[t + o];
            __syncthreads();
        }
        if (t == 0) ytr[ij] = red[0];
        __syncthreads();
    }

    if (t == 0) {
        float tot = 0.0f;
        for (int ij = 0; ij < IJ; ++ij) tot += ytr[ij];
        float itot = 1.0f / tot;
        float mr[2][2];
        mr[0][0] = (ytr[0] + ytr[1] + ytr[3] + ytr[4]) * itot; // both observed
        mr[0][1] = (ytr[2] + ytr[5]) * itot;                   // r missing
        mr[1][0] = (ytr[6] + ytr[7]) * itot;                   // t missing
        mr[1][1] = ytr[8] * itot;                              // both missing
        float* Lmr = ws + WS_LMR;
        for (int i = 0; i < 3; ++i)
            for (int j = 0; j < 3; ++j)
                Lmr[(i * 3 + j) * B_DIM + b] = __logf(mr[i == 2][j == 2]);
    }
}

// ---------------------------------------------------------------------------
// Fuse the b-independent weights: W[ijenc][c] = logT[ij][c] + Lq[ijenc][c%20]
__global__ void build_W(float* __restrict__ ws) {
    const float* logT = ws + WS_LOGT;
    const float* Lq   = ws + WS_LQ;
    float* W          = ws + WS_W;
    const int blk = blockIdx.x;        // ij*16 + enc
    const int ij  = blk >> 4;
    for (int c = threadIdx.x; c < VK; c += blockDim.x) {
        int k = c % K_DIM;
        W[(size_t)blk * VK + c] = logT[ij * VK + c] + Lq[blk * K_DIM + k];
    }
}

// ---------------------------------------------------------------------------
// Main streaming contraction. One wave per 16-row tile of yphi.
// D = A(yphi 16x4) x B(W chunk, replicated over columns) + C, accumulated over
// the 480 K=4 chunks with V_WMMA_F32_16X16X4_F32. Row totals carry the Lmr term.
__global__ void main_reduce(const float* __restrict__ yphi,
                            const float* __restrict__ ws,
                            float* __restrict__ out) {
    const float* W   = ws + WS_W;
    const float* Lmr = ws + WS_LMR;

    const int tile  = blockIdx.x;       // 0..2303
    const int bt    = tile & 15;        // 16-row tile within B
    const int ijenc = tile >> 4;        // 0..143
    const int ij    = ijenc >> 4;

    const int l   = threadIdx.x;        // 0..31, wave32
    const int m   = l & 15;             // A-matrix row (b within tile)
    const int off = (l >> 4) * 2;       // K offset within chunk: 0 or 2
    const int row = bt * 16 + m;        // b index

    const float* yrow = yphi + (size_t)(ijenc * B_DIM + row) * VK + off;
    const float* wrow = W    + (size_t)ijenc * VK + off;

    v8f acc;
#pragma unroll
    for (int i = 0; i < 8; ++i) acc[i] = 0.0f;
    float rowtot = 0.0f;

#pragma unroll 4
    for (int c0 = 0; c0 < VK; c0 += 4) {
        v2f a  = *(const v2f*)(yrow + c0);   // global_load_b64: A layout exactly
        v2f bb = *(const v2f*)(wrow + c0);   // replicated-column B operand
        rowtot += a.x + a.y;
        acc = __builtin_amdgcn_wmma_f32_16x16x4_f32(
            /*neg_a=*/false, a, /*neg_b=*/false, bb,
            /*c_mod=*/(short)0, acc, /*reuse_a=*/false, /*reuse_b=*/false);
    }

    // Each D column repeats the row-dot 16x -> divide lane sum by 16.
    float q = 0.0f;
#pragma unroll
    for (int i = 0; i < 8; ++i) q += acc[i];
    q *= (1.0f / 16.0f);
    q += rowtot * Lmr[ij * B_DIM + row];

    // wave32 reduction
    for (int s = 16; s > 0; s >>= 1) q += __shfl_xor(q, s, 32);

    if (l == 0) atomicAdd(out, -q * (1.0f / (float)VK));
}

// ---------------------------------------------------------------------------
extern "C" void kernel_launch(void* const* d_in, const int* in_sizes, int n_in,
                              void* d_out, int out_size, void* d_ws, size_t ws_size,
                              hipStream_t stream) {
    const float* yphi = (const float*)d_in[0];
    const float* Y    = (const float*)d_in[1];
    const float* T0   = (const float*)d_in[2];
    const float* t_   = (const float*)d_in[3];
    const float* r_   = (const float*)d_in[4];
    const float* e_   = (const float*)d_in[5];
    const float* n_   = (const float*)d_in[6];
    const float* c_   = (const float*)d_in[7];
    const int*   idx  = (const int*)d_in[8];
    float* ws  = (float*)d_ws;
    float* out = (float*)d_out;

    zero_out  <<<1, 32, 0, stream>>>(out);
    prep_small<<<1, 32, 0, stream>>>(T0, t_, r_, e_, n_, c_, ws);
    prep_mr   <<<B_DIM, 256, 0, stream>>>(Y, idx, ws);
    build_W   <<<IJ * ENC, 256, 0, stream>>>(ws);
    main_reduce<<<IJ * ENC * (B_DIM / 16), 32, 0, stream>>>(yphi, ws, out);
}